// Model_22926535426081
// MI455X (gfx1250) — compile-verified
//
#include <hip/hip_runtime.h>
#include <math.h>

#define B_ 256
#define S_ 336
#define F_ 8
#define H_ 512
#define P_ 96
#define G_ (4 * H_)   // 2048 gate columns
#define KC 128        // K-chunk staged in LDS

typedef __attribute__((ext_vector_type(2))) float v2f;
typedef __attribute__((ext_vector_type(8))) float v8f;

__device__ __forceinline__ float sigf(float x) { return 1.0f / (1.0f + expf(-x)); }

// -------- one-time weight packing -------------------------------------------
// Pack W (G_ x K row-major) into WMMA-B-fragment-ready layout:
//   Wpk[(((q*2 + khi)*G_ + g)*4) + j] = W[g*K + k]
//   with k = 8*q + 4*(j>>1) + 2*khi + (j&1),  q = K-octet index.
// In the GEMM, lane (khi,g) then fetches its four weights for one K-octet
// with a single coalesced global_load_b128.
__global__ void packB_kernel(const float* __restrict__ src,
                             float* __restrict__ dst, int K) {
  int idx = blockIdx.x * blockDim.x + threadIdx.x;
  if (idx >= K * G_) return;
  int j   = idx & 3;
  int g   = (idx >> 2) % G_;
  int qk  = (idx >> 2) / G_;     // q*2 + khi
  int q   = qk >> 1, khi = qk & 1;
  int k   = 8 * q + 4 * (j >> 1) + 2 * khi + (j & 1);
  dst[idx] = src[(size_t)g * K + k];
}

// x_dec0[b] = x_enc[b, S-1, 3]
__global__ void init_xdec_kernel(const float* __restrict__ x_enc,
                                 float* __restrict__ xdec) {
  int b = blockIdx.x * blockDim.x + threadIdx.x;
  if (b < B_) xdec[b] = x_enc[(size_t)b * S_ * F_ + (S_ - 1) * F_ + 3];
}

// -------- fused per-timestep LSTM kernel ------------------------------------
// grid = (8 batch-tiles of 32, 8 col-tiles of 64); block = 128 (4 waves).
// wave w computes gate region w: gates[:, w*512 + j0 .. +64) via f32 WMMA.
__global__ __launch_bounds__(128)
void lstm_step_kernel(int mode,                              // 0=encoder 1=decoder
                      const float* __restrict__ h_in,
                      float* __restrict__ h_out,
                      float* __restrict__ c_io,
                      const float* __restrict__ Wpk,         // packed 512xG_
                      const float* __restrict__ bias,        // 2048
                      const float* __restrict__ x_enc, int t,
                      const float* __restrict__ Wpk_ih,      // packed 8xG_ (mode 0)
                      const float* __restrict__ x_dec,       // B (mode 1)
                      const float* __restrict__ Wih_d) {     // 2048 (mode 1)
  __shared__ __align__(16) float hA[32][KC + 4];   // padded: conflict-free b64 A-frags
  __shared__ __align__(16) float xA[32][8];
  __shared__ __align__(16) float gsh[4][32][65];   // staged gate tiles (i,f,g,o)

  const int tid  = threadIdx.x;
  const int wave = tid >> 5;            // gate index 0..3
  const int lane = tid & 31;
  const int n16  = lane & 15;
  const int khi  = lane >> 4;           // which K-pair this half-wave holds
  const int row0 = blockIdx.x * 32;     // batch rows
  const int j0   = blockIdx.y * 64;     // hidden cols
  const int g0   = wave * H_ + j0;      // column base in the 2048-wide gate matrix

  v8f acc[2][4];
  {
    v8f z;
#pragma unroll
    for (int i = 0; i < 8; ++i) z[i] = 0.0f;
#pragma unroll
    for (int mt = 0; mt < 2; ++mt)
#pragma unroll
      for (int nt = 0; nt < 4; ++nt) acc[mt][nt] = z;
  }

  // stage x tile
  if (mode == 0) {
    for (int i = tid; i < 32 * 8; i += 128) {
      int m = i >> 3, f = i & 7;
      xA[m][f] = x_enc[(size_t)(row0 + m) * S_ * F_ + (size_t)t * F_ + f];
    }
  } else {
    if (tid < 32) xA[tid][0] = x_dec[row0 + tid];
  }

  // ---- main GEMM: gates += h_in @ Whh^T over K = 512 in chunks of KC ------
  for (int kc = 0; kc < H_; kc += KC) {
    __syncthreads();
    {  // cooperative load: 32 rows x KC floats of h into LDS
      int r = tid >> 2, q = tid & 3;
      const float4* src = (const float4*)(h_in + (size_t)(row0 + r) * H_ + kc);
      float4* dstrow = (float4*)&hA[r][0];
#pragma unroll
      for (int i = 0; i < 8; ++i) dstrow[q + 4 * i] = src[q + 4 * i];
    }
    __syncthreads();

    for (int kk = 0; kk < KC; kk += 8) {     // one K-octet per iteration
      const int q = (kc + kk) >> 3;
      v2f a0lo = *(const v2f*)&hA[n16][kk + 2 * khi];
      v2f a1lo = *(const v2f*)&hA[16 + n16][kk + 2 * khi];
      v2f a0hi = *(const v2f*)&hA[n16][kk + 4 + 2 * khi];
      v2f a1hi = *(const v2f*)&hA[16 + n16][kk + 4 + 2 * khi];

      float4 bq[4];
#pragma unroll
      for (int nt = 0; nt < 4; ++nt) {       // one b128 per N-tile per octet
        const int g = g0 + nt * 16 + n16;
        bq[nt] = *(const float4*)&Wpk[(((size_t)q * 2 + khi) * G_ + g) * 4];
      }
#pragma unroll
      for (int nt = 0; nt < 4; ++nt) {
        v2f blo; blo.x = bq[nt].x; blo.y = bq[nt].y;
        acc[0][nt] = __builtin_amdgcn_wmma_f32_16x16x4_f32(
            false, a0lo, false, blo, (short)0, acc[0][nt], false, false);
        acc[1][nt] = __builtin_amdgcn_wmma_f32_16x16x4_f32(
            false, a1lo, false, blo, (short)0, acc[1][nt], false, false);
      }
#pragma unroll
      for (int nt = 0; nt < 4; ++nt) {
        v2f bhi; bhi.x = bq[nt].z; bhi.y = bq[nt].w;
        acc[0][nt] = __builtin_amdgcn_wmma_f32_16x16x4_f32(
            false, a0hi, false, bhi, (short)0, acc[0][nt], false, false);
        acc[1][nt] = __builtin_amdgcn_wmma_f32_16x16x4_f32(
            false, a1hi, false, bhi, (short)0, acc[1][nt], false, false);
      }
    }
  }

  // ---- input contribution ------------------------------------------------
  if (mode == 0) {
    // one packed K-octet: x(32x8) @ Wih^T
    v2f a0lo = *(const v2f*)&xA[n16][2 * khi];
    v2f a1lo = *(const v2f*)&xA[16 + n16][2 * khi];
    v2f a0hi = *(const v2f*)&xA[n16][4 + 2 * khi];
    v2f a1hi = *(const v2f*)&xA[16 + n16][4 + 2 * khi];
#pragma unroll
    for (int nt = 0; nt < 4; ++nt) {
      const int g = g0 + nt * 16 + n16;
      float4 bq = *(const float4*)&Wpk_ih[((size_t)khi * G_ + g) * 4];
      v2f blo; blo.x = bq.x; blo.y = bq.y;
      v2f bhi; bhi.x = bq.z; bhi.y = bq.w;
      acc[0][nt] = __builtin_amdgcn_wmma_f32_16x16x4_f32(
          false, a0lo, false, blo, (short)0, acc[0][nt], false, false);
      acc[1][nt] = __builtin_amdgcn_wmma_f32_16x16x4_f32(
          false, a1lo, false, blo, (short)0, acc[1][nt], false, false);
      acc[0][nt] = __builtin_amdgcn_wmma_f32_16x16x4_f32(
          false, a0hi, false, bhi, (short)0, acc[0][nt], false, false);
      acc[1][nt] = __builtin_amdgcn_wmma_f32_16x16x4_f32(
          false, a1hi, false, bhi, (short)0, acc[1][nt], false, false);
    }
  } else {
    // decoder input is K=1: scalar FMA
#pragma unroll
    for (int mt = 0; mt < 2; ++mt)
#pragma unroll
      for (int nt = 0; nt < 4; ++nt) {
        float w = Wih_d[g0 + nt * 16 + n16];
#pragma unroll
        for (int v = 0; v < 8; ++v)
          acc[mt][nt][v] += xA[mt * 16 + 8 * khi + v][0] * w;
      }
  }

  // ---- bias + stage gates to LDS ------------------------------------------
#pragma unroll
  for (int nt = 0; nt < 4; ++nt) {
    float bb = bias[g0 + nt * 16 + n16];
#pragma unroll
    for (int mt = 0; mt < 2; ++mt)
#pragma unroll
      for (int v = 0; v < 8; ++v) {
        float val = acc[mt][nt][v] + bb;
        gsh[wave][mt * 16 + 8 * khi + v][nt * 16 + n16] = val;
      }
  }
  __syncthreads();

  // ---- fused LSTM elementwise update --------------------------------------
  for (int idx = tid; idx < 32 * 64; idx += 128) {
    int m = idx >> 6, j = idx & 63;
    float gi = gsh[0][m][j];
    float gf = gsh[1][m][j];
    float gg = gsh[2][m][j];
    float go = gsh[3][m][j];
    size_t off = (size_t)(row0 + m) * H_ + j0 + j;
    float c  = c_io[off];
    float cn = sigf(gf) * c + sigf(gi) * tanhf(gg);
    float hn = sigf(go) * tanhf(cn);
    c_io[off]  = cn;
    h_out[off] = hn;
  }
}

// -------- decoder output projection: y = h @ lin_W.T + lin_b ----------------
__global__ __launch_bounds__(128)
void dec_out_kernel(const float* __restrict__ h, const float* __restrict__ linW,
                    const float* __restrict__ linb, float* __restrict__ out,
                    float* __restrict__ xdec, int p) {
  __shared__ float red[128];
  int b = blockIdx.x, tid = threadIdx.x;
  const float4* hp = (const float4*)(h + (size_t)b * H_);
  const float4* wp = (const float4*)linW;
  float4 hv = hp[tid];
  float4 wv = wp[tid];
  red[tid] = hv.x * wv.x + hv.y * wv.y + hv.z * wv.z + hv.w * wv.w;
  __syncthreads();
  for (int off = 64; off > 0; off >>= 1) {
    if (tid < off) red[tid] += red[tid + off];
    __syncthreads();
  }
  if (tid == 0) {
    float y = red[0] + linb[0];
    out[(size_t)b * P_ + p] = y;
    xdec[b] = y;
  }
}

// ---------------------------------------------------------------------------

extern "C" void kernel_launch(void* const* d_in, const int* in_sizes, int n_in,
                              void* d_out, int out_size, void* d_ws, size_t ws_size,
                              hipStream_t stream) {
  const float* x_enc = (const float*)d_in[0];
  const float* eWih  = (const float*)d_in[1];  // (2048, 8)
  const float* eWhh  = (const float*)d_in[2];  // (2048, 512)
  const float* eb    = (const float*)d_in[3];  // (2048,)
  const float* dWih  = (const float*)d_in[4];  // (2048, 1)
  const float* dWhh  = (const float*)d_in[5];  // (2048, 512)
  const float* db    = (const float*)d_in[6];  // (2048,)
  const float* linW  = (const float*)d_in[7];  // (1, 512)
  const float* linb  = (const float*)d_in[8];  // (1,)
  float* out = (float*)d_out;

  float* p = (float*)d_ws;
  float* Wpk_e  = p; p += 512 * G_;
  float* Wpk_d  = p; p += 512 * G_;
  float* Wpk_ih = p; p += 8 * G_;
  float* hA   = p; p += B_ * H_;
  float* hB   = p; p += B_ * H_;
  float* cbuf = p; p += B_ * H_;
  float* xdec = p; p += B_;

  // one-time weight packing into WMMA-B-fragment order (L2-resident afterwards)
  packB_kernel<<<(512 * G_ + 255) / 256, 256, 0, stream>>>(eWhh, Wpk_e, 512);
  packB_kernel<<<(512 * G_ + 255) / 256, 256, 0, stream>>>(dWhh, Wpk_d, 512);
  packB_kernel<<<(8 * G_ + 255) / 256, 256, 0, stream>>>(eWih, Wpk_ih, 8);
  hipMemsetAsync(hA, 0, (size_t)B_ * H_ * sizeof(float), stream);
  hipMemsetAsync(cbuf, 0, (size_t)B_ * H_ * sizeof(float), stream);

  dim3 grid(8, 8), block(128);
  float* hin = hA;
  float* hout = hB;

  for (int t = 0; t < S_; ++t) {
    lstm_step_kernel<<<grid, block, 0, stream>>>(
        0, hin, hout, cbuf, Wpk_e, eb, x_enc, t, Wpk_ih, nullptr, nullptr);
    float* tmp = hin; hin = hout; hout = tmp;
  }

  init_xdec_kernel<<<1, 256, 0, stream>>>(x_enc, xdec);

  for (int ps = 0; ps < P_; ++ps) {
    lstm_step_kernel<<<grid, block, 0, stream>>>(
        1, hin, hout, cbuf, Wpk_d, db, nullptr, 0, nullptr, xdec, dWih);
    float* tmp = hin; hin = hout; hout = tmp;
    dec_out_kernel<<<B_, 128, 0, stream>>>(hin, linW, linb, out, xdec, ps);
  }
}